// SoftSearch_9199819948193
// MI455X (gfx1250) — compile-verified
//
#include <hip/hip_runtime.h>
#include <hip/hip_bf16.h>
#include <math.h>

typedef __attribute__((ext_vector_type(2))) float v2f;
typedef __attribute__((ext_vector_type(8))) float v8f;

#define SS_B 1024
#define SS_S 4096
#define SS_D 64
#define SS_NT (SS_S / 16)   // 256 tiles of 16 rows
#define SS_EPS 1e-8f

// One block per batch row b. 256 threads = 8 wave32.
// Phase 1: WMMA f32 16x16x4 computes dot(candidate, behavior_row) for 16 rows
//          per tile; a second WMMA with (A,A) computes the Gram tile whose
//          diagonal is sum(x^2) per row -> row norms for free.
//          Manual 2-stage software pipeline with statically-indexed register
//          buffers (no dynamic indexing -> no scratch spill).
// Phase 2: iterative stable arg-max selection of top-k from LDS.
__global__ void __launch_bounds__(256)
softsearch_topk_kernel(const float* __restrict__ cand,
                       const float* __restrict__ beh,
                       int* __restrict__ out,
                       int k) {
    __shared__ float s_sim[SS_S];      // 16 KB
    __shared__ float s_rval[8];
    __shared__ int   s_ridx[8];
    __shared__ int   s_win;

    const int b    = blockIdx.x;
    const int tid  = threadIdx.x;
    const int lane = tid & 31;
    const int wave = tid >> 5;

    // ---------------- Phase 1: similarities via WMMA ----------------
    // K-mapping per WMMA f32 16x16x4 fragment layout:
    // lanes 0-15 hold K = kc+0, kc+1 ; lanes 16-31 hold K = kc+2, kc+3
    const int co  = (lane >> 4) << 1;          // 0 or 2
    const int row = lane & 15;                 // M row within tile

    const float* crow = cand + (size_t)b * SS_D;

    // Candidate B fragments (broadcast across all 16 N columns) + cand norm.
    v2f bfrag[16];
    float css = 0.f;
#pragma unroll
    for (int kc = 0; kc < 16; ++kc) {
        float x0 = crow[kc * 4 + co];
        float x1 = crow[kc * 4 + co + 1];
        v2f t; t.x = x0; t.y = x1;
        bfrag[kc] = t;
        css += x0 * x0 + x1 * x1;
    }
    // lanes 0-15 cover K halves {0,1 mod 4}, lanes 16-31 cover {2,3 mod 4}
    css += __shfl_xor(css, 16, 32);
    const float cn = fmaxf(sqrtf(css), SS_EPS);

    const float* bb    = beh + (size_t)b * SS_S * SS_D;
    const int    idx   = lane & 7;
    const size_t lbase = (size_t)row * SS_D + co;

    // Load one tile's 16 A fragments into a statically-indexed register array.
    auto load_tile = [&](int t, v2f (&a)[16]) {
        const float* arow = bb + (size_t)(t * 16) * SS_D + lbase;
#pragma unroll
        for (int kc = 0; kc < 16; ++kc)
            a[kc] = *(const v2f*)(arow + kc * 4);
    };

    // 32 WMMAs + norm/sim extraction + LDS store for one tile.
    auto compute_tile = [&](int t, const v2f (&a)[16]) {
        v8f c = {};   // dot tile (all 16 columns identical)
        v8f g = {};   // Gram tile A*A^T (diagonal = sum of squares)
#pragma unroll
        for (int kc = 0; kc < 16; ++kc) {
            c = __builtin_amdgcn_wmma_f32_16x16x4_f32(false, a[kc], false, bfrag[kc],
                                                      (short)0, c, false, false);
            g = __builtin_amdgcn_wmma_f32_16x16x4_f32(false, a[kc], false, a[kc],
                                                      (short)0, g, false, false);
        }
        // Diagonal & matching dot both live in-lane:
        //   lanes 0-7  : row r = lane      -> c[lane],    g[lane]
        //   lanes 24-31: row r = lane - 16 -> c[lane-24], g[lane-24]
        float dot = 0.f, ss = 0.f;
#pragma unroll
        for (int r = 0; r < 8; ++r) {
            if (r == idx) { dot = c[r]; ss = g[r]; }
        }
        const float bn  = fmaxf(sqrtf(ss), SS_EPS);
        const float sim = dot / (cn * bn);

        const int s0 = t * 16;
        if (lane < 8)        s_sim[s0 + lane]      = sim;
        else if (lane >= 24) s_sim[s0 + lane - 16] = sim;
    };

    // Manual 2-stage pipeline; A0/A1 only ever statically indexed.
    v2f A0[16], A1[16];
    load_tile(wave, A0);
    for (int t = wave; t < SS_NT; t += 16) {
        load_tile(t + 8, A1);          // in flight while computing A0
        compute_tile(t, A0);
        if (t + 16 < SS_NT) load_tile(t + 16, A0);  // in flight while computing A1
        compute_tile(t + 8, A1);
    }
    __syncthreads();

    // ---------------- Phase 2: stable top-k selection ----------------
    // Thread tid owns elements s = tid + j*256, j = 0..15 (registers).
    float v[16];
    unsigned alive = 0xFFFFu;
#pragma unroll
    for (int j = 0; j < 16; ++j) v[j] = s_sim[tid + j * 256];

    for (int it = 0; it < k; ++it) {
        float best = -INFINITY;
        int   bi   = 0x7FFFFFFF;
#pragma unroll
        for (int j = 0; j < 16; ++j) {
            if ((alive >> j) & 1u) {
                float vv = v[j];
                if (vv > best) { best = vv; bi = tid + j * 256; }  // keeps lowest s
            }
        }
        // wave32 reduction: max value, tie -> lower index
#pragma unroll
        for (int off = 16; off > 0; off >>= 1) {
            float ov = __shfl_down(best, off, 32);
            int   oi = __shfl_down(bi,   off, 32);
            if (ov > best || (ov == best && oi < bi)) { best = ov; bi = oi; }
        }
        if (lane == 0) { s_rval[wave] = best; s_ridx[wave] = bi; }
        __syncthreads();

        if (tid == 0) {
            float wb = s_rval[0]; int wi = s_ridx[0];
#pragma unroll
            for (int w = 1; w < 8; ++w) {
                float ov = s_rval[w]; int oi = s_ridx[w];
                if (ov > wb || (ov == wb && oi < wi)) { wb = ov; wi = oi; }
            }
            s_win = wi;
            out[(size_t)b * k + it] = wi;
        }
        __syncthreads();

        const int wi = s_win;
        if ((wi & 255) == tid) alive &= ~(1u << (wi >> 8));
        // next write to s_win is after the first barrier of the next
        // iteration, so no extra barrier needed here
    }
}

extern "C" void kernel_launch(void* const* d_in, const int* in_sizes, int n_in,
                              void* d_out, int out_size, void* d_ws, size_t ws_size,
                              hipStream_t stream) {
    (void)in_sizes; (void)n_in; (void)d_ws; (void)ws_size;
    const float* cand = (const float*)d_in[0];   // (1024, 64) f32
    const float* beh  = (const float*)d_in[1];   // (1024, 4096, 64) f32
    int* out = (int*)d_out;                      // (1024, k) int32 indices
    const int k = out_size / SS_B;               // k=100; device scalar d_in[2] unreadable on host

    softsearch_topk_kernel<<<SS_B, 256, 0, stream>>>(cand, beh, out, k);
}